// DTL_54743653154988
// MI455X (gfx1250) — compile-verified
//
#include <hip/hip_runtime.h>
#include <stdint.h>

#define BDIM 256
#define DELTA_W 0.2f

// Map float bit pattern to a monotonically ordered unsigned key.
__device__ __forceinline__ unsigned f2k(unsigned u) {
    return (u & 0x80000000u) ? ~u : (u | 0x80000000u);
}
__device__ __forceinline__ float k2f(unsigned k) {
    unsigned u = (k & 0x80000000u) ? (k ^ 0x80000000u) : ~k;
    union { unsigned u; float f; } c; c.u = u; return c.f;
}

// One workgroup per row:
//   1) async-copy row into LDS (bulk GLOBAL_LOAD_ASYNC_TO_LDS_B128, b32 head/tail)
//   2) mask positive slot with -inf (exact reference semantics), save pos
//   3) 4-pass byte-radix select over LDS (uint4 scans) -> exact K-th key + ties
//   4) accumulate (1+x)^2 over top-K, combine with positive term
__global__ __launch_bounds__(BDIM)
void dtl_row_kernel(const float* __restrict__ inputs,
                    const int*   __restrict__ targets,
                    float*       __restrict__ partial,
                    int n, int histoff, int K) {
    extern __shared__ __align__(16) unsigned char smem[];
    unsigned* hist = (unsigned*)(smem + (size_t)histoff);            // 256 u32
    unsigned* gtot = (unsigned*)(smem + (size_t)histoff + 1024u);    // 8 u32 (reused as float red)
    float*    red  = (float*)gtot;
    unsigned* ctrl = (unsigned*)(smem + (size_t)histoff + 2048u);    // 3 u32

    const int tid = threadIdx.x;
    const int row = blockIdx.x;
    const int tgt = targets[row];

    // ---- Phase 1: async global -> LDS copy of this row (CDNA5 async path) ----
    const uint64_t gbase  = (uint64_t)(uintptr_t)(inputs + (size_t)row * (size_t)n);
    const unsigned s      = (unsigned)(gbase & 15u);          // 0,4,8,12 (row is 4B aligned)
    const uint32_t lbase0 = (uint32_t)(uintptr_t)smem;        // low 32 bits == LDS byte offset
    float* srow = (float*)(smem + s);                         // element j lives at smem + s + 4j

    int h = (int)(((16u - s) & 15u) >> 2);                    // head elems before 16B alignment
    if (h > n) h = n;
    const int nb    = (n - h) >> 2;                           // # of 16B chunks (copy AND scan)
    const int tail0 = h + (nb << 2);
    const int tcnt  = n - tail0;

    for (int c = tid; c < nb; c += BDIM) {                    // bulk: 16B per lane per issue
        uint32_t goff  = (uint32_t)((h << 2) + (c << 4));
        uint32_t laddr = lbase0 + s + goff;
        asm volatile("global_load_async_to_lds_b128 %0, %1, %2"
                     :: "v"(laddr), "v"(goff), "s"(gbase) : "memory");
    }
    if (tid < h) {                                            // head (<=3 elems)
        uint32_t goff  = (uint32_t)(tid << 2);
        uint32_t laddr = lbase0 + s + goff;
        asm volatile("global_load_async_to_lds_b32 %0, %1, %2"
                     :: "v"(laddr), "v"(goff), "s"(gbase) : "memory");
    }
    if (tid < tcnt) {                                         // tail (<=3 elems)
        uint32_t goff  = (uint32_t)((tail0 + tid) << 2);
        uint32_t laddr = lbase0 + s + goff;
        asm volatile("global_load_async_to_lds_b32 %0, %1, %2"
                     :: "v"(laddr), "v"(goff), "s"(gbase) : "memory");
    }
    asm volatile("s_wait_asynccnt 0" ::: "memory");
    __syncthreads();

    // ---- Save positive logit, overwrite its slot with -inf (== reference mask) ----
    if (tid == 0) {
        ctrl[2]   = __float_as_uint(srow[tgt]);
        srow[tgt] = __uint_as_float(0xFF800000u);             // -inf
    }
    __syncthreads();

    const uint4* srow4 = (const uint4*)(smem + s + ((unsigned)h << 2));   // 16B aligned
    const int lane = tid & 31;
    const int wave = tid >> 5;

    // ---- Phase 2: radix select the K-th largest key (exact, incl. ties) ----
    unsigned prefix    = 0u;
    unsigned remaining = (unsigned)K;
#pragma unroll
    for (int shift = 24; shift >= 0; shift -= 8) {
        const unsigned hiMask = (shift == 24) ? 0u : (0xFFFFFFFFu << (shift + 8));
        hist[tid] = 0u;
        __syncthreads();

        for (int c = tid; c < nb; c += BDIM) {                // 4 elems per ds_load_b128
            uint4 v = srow4[c];
            unsigned k0 = f2k(v.x); if ((k0 & hiMask) == prefix) atomicAdd(&hist[(k0 >> shift) & 0xFFu], 1u);
            unsigned k1 = f2k(v.y); if ((k1 & hiMask) == prefix) atomicAdd(&hist[(k1 >> shift) & 0xFFu], 1u);
            unsigned k2 = f2k(v.z); if ((k2 & hiMask) == prefix) atomicAdd(&hist[(k2 >> shift) & 0xFFu], 1u);
            unsigned k3 = f2k(v.w); if ((k3 & hiMask) == prefix) atomicAdd(&hist[(k3 >> shift) & 0xFFu], 1u);
        }
        if (tid < h) {
            unsigned k = f2k(__float_as_uint(srow[tid]));
            if ((k & hiMask) == prefix) atomicAdd(&hist[(k >> shift) & 0xFFu], 1u);
        }
        if (tid < tcnt) {
            unsigned k = f2k(__float_as_uint(srow[tail0 + tid]));
            if ((k & hiMask) == prefix) atomicAdd(&hist[(k >> shift) & 0xFFu], 1u);
        }
        __syncthreads();

        // Wave-level inclusive suffix-sum over this wave's 32 bins (no barriers).
        const unsigned cnt = hist[tid];
        unsigned val = cnt;
#pragma unroll
        for (int d = 1; d < 32; d <<= 1) {
            unsigned t = __shfl_down(val, (unsigned)d, 32);
            if (lane + d < 32) val += t;
        }
        if (lane == 0) gtot[wave] = val;                      // total of bins [32w, 32w+31]
        __syncthreads();
        unsigned gsum = 0u;
#pragma unroll
        for (int g = 0; g < 8; ++g)
            if (g > wave) gsum += gtot[g];
        const unsigned ge = val + gsum;                       // #keys with digit >= tid
        const unsigned gt = ge - cnt;                         // #keys with digit >  tid
        if (ge >= remaining && gt < remaining) {              // unique winner writes ctrl
            ctrl[0] = prefix | ((unsigned)tid << shift);
            ctrl[1] = remaining - gt;
        }
        __syncthreads();
        prefix    = ctrl[0];
        remaining = ctrl[1];
        __syncthreads();
    }
    const unsigned kthKey = prefix;          // exact K-th largest key
    const unsigned nTies  = remaining;       // elements equal to kthKey to include

    // ---- Phase 3: accumulate (1 + x)^2 over strictly-greater keys ----
    float acc = 0.0f;
    for (int c = tid; c < nb; c += BDIM) {
        uint4 v = srow4[c];
        if (f2k(v.x) > kthKey) { float t = 1.0f + __uint_as_float(v.x); acc += t * t; }
        if (f2k(v.y) > kthKey) { float t = 1.0f + __uint_as_float(v.y); acc += t * t; }
        if (f2k(v.z) > kthKey) { float t = 1.0f + __uint_as_float(v.z); acc += t * t; }
        if (f2k(v.w) > kthKey) { float t = 1.0f + __uint_as_float(v.w); acc += t * t; }
    }
    if (tid < h) {
        unsigned u = __float_as_uint(srow[tid]);
        if (f2k(u) > kthKey) { float t = 1.0f + __uint_as_float(u); acc += t * t; }
    }
    if (tid < tcnt) {
        unsigned u = __float_as_uint(srow[tail0 + tid]);
        if (f2k(u) > kthKey) { float t = 1.0f + __uint_as_float(u); acc += t * t; }
    }
#pragma unroll
    for (int d = 16; d > 0; d >>= 1)
        acc += __shfl_down(acc, (unsigned)d, 32);
    if (lane == 0) red[wave] = acc;
    __syncthreads();
    if (tid == 0) {
        float sum = 0.0f;
#pragma unroll
        for (int g = 0; g < 8; ++g) sum += red[g];
        float tk      = 1.0f + k2f(kthKey);
        float hardsum = sum + (float)nTies * tk * tk;
        float pos     = __uint_as_float(ctrl[2]);
        float pr      = (1.0f - pos) * (1.0f - pos) + DELTA_W * (hardsum / (float)K);
        partial[row]  = pr;
    }
}

// Deterministic final mean over the per-row partials.
__global__ __launch_bounds__(BDIM)
void dtl_mean_kernel(const float* __restrict__ partial, float* __restrict__ out, int m) {
    __shared__ float red[BDIM];
    float a = 0.0f;
    for (int i = threadIdx.x; i < m; i += BDIM) a += partial[i];
    red[threadIdx.x] = a;
    __syncthreads();
    for (int s = BDIM / 2; s > 0; s >>= 1) {
        if (threadIdx.x < s) red[threadIdx.x] += red[threadIdx.x + s];
        __syncthreads();
    }
    if (threadIdx.x == 0) out[0] = red[0] / (float)m;
}

extern "C" void kernel_launch(void* const* d_in, const int* in_sizes, int n_in,
                              void* d_out, int out_size, void* d_ws, size_t ws_size,
                              hipStream_t stream) {
    const float* inputs  = (const float*)d_in[0];
    const int*   targets = (const int*)d_in[1];
    float*       out     = (float*)d_out;
    float*       partial = (float*)d_ws;

    const int m = in_sizes[1];                       // 4096 rows
    const int n = in_sizes[0] / m;                   // 10001 classes
    int K = (int)(0.01 * (double)(n - 1));           // int(R*(n-1)) == 100
    if (K < 1) K = 1;

    // Row region: n floats + up to 12B alignment shift, rounded to 16B, +16 margin.
    const int    histoff = (((n * 4) + 15) & ~15) + 16;
    const size_t smem    = (size_t)histoff + 1024u + 1024u + 16u;  // ~42 KB

    dtl_row_kernel<<<m, BDIM, smem, stream>>>(inputs, targets, partial, n, histoff, K);
    dtl_mean_kernel<<<1, BDIM, 0, stream>>>(partial, out, m);
}